// CausalSelfAttention_82643760710012
// MI455X (gfx1250) — compile-verified
//
#include <hip/hip_runtime.h>
#include <hip/hip_bf16.h>

// ---------------------------------------------------------------------------
// CDNA5 (gfx1250) causal self-attention, fp32 via V_WMMA_F32_16X16X4_F32,
// with double-buffered GLOBAL_LOAD_ASYNC_TO_LDS_B128 tile staging (ASYNCcnt).
// ---------------------------------------------------------------------------

typedef float v2f __attribute__((ext_vector_type(2)));
typedef float v8f __attribute__((ext_vector_type(8)));

#define NEG_INF (-__builtin_inff())

// D = A(16x4 f32) * B(4x16 f32) + C(16x16 f32)
__device__ __forceinline__ v8f wmma4(v2f a, v2f b, v8f c) {
  return __builtin_amdgcn_wmma_f32_16x16x4_f32(
      /*neg_a=*/false, a, /*neg_b=*/false, b,
      /*c_mod=*/(short)0, c, /*reuse_a=*/false, /*reuse_b=*/false);
}

// Async global->LDS 16B copy (per lane), GVS mode: saddr(u64) + voffset(u32).
// Generic LDS pointer low 32 bits == LDS byte offset (ISA: LDS aperture
// truncates to addr[31:0]).
__device__ __forceinline__ void async_ld_b128(void* lds, const void* gbase,
                                              unsigned byteOff) {
  unsigned l = (unsigned)(uintptr_t)lds;
  asm volatile("global_load_async_to_lds_b128 %0, %1, %2"
               :
               : "v"(l), "v"(byteOff), "s"(gbase)
               : "memory");
}
__device__ __forceinline__ void wait_async0() {
  asm volatile("s_wait_asynccnt 0" ::: "memory");
}

// ===========================================================================
// Kernel 1: C[M,N] = A[M,K] @ B[K,N] + bias[N]     (row-major, fp32)
// 256 threads = 8 waves; block tile 128x64; wave tile 32x32 (2x2 WMMA tiles);
// K chunk = 16. Double-buffered async LDS staging; padded strides (20 / 80)
// keep both b128 staging 16B-aligned and fragment reads bank-conflict-free.
// ===========================================================================
__global__ __launch_bounds__(256) void gemm_bias_wmma(
    const float* __restrict__ A, const float* __restrict__ Bm,
    const float* __restrict__ bias, float* __restrict__ C,
    int M, int N, int K) {
  __shared__ float As[2][128 * 20];
  __shared__ float Bs[2][16 * 80];

  const int tid  = threadIdx.x;
  const int lane = tid & 31;
  const int wid  = tid >> 5;
  const int wRow = (wid & 3) * 32;        // 0,32,64,96
  const int wCol = (wid >> 2) * 32;       // 0,32
  const int lh   = lane & 15;
  const int half = lane >> 4;
  const int selk = half * 2;              // K-split across half-waves
  const int rowBase = blockIdx.y * 128;
  const int colBase = blockIdx.x * 64;

  // staging coords: A-tile 128x16 (2 x b128/thread), B-tile 16x64 (1 x b128)
  const int aRow = tid >> 1;
  const int aCol = (tid & 1) * 8;
  const int bRow = tid >> 4;
  const int bCol = (tid & 15) * 4;
  const int nck = K >> 4;

  v8f acc[2][2] = {};

  // prologue: stage chunk 0 into buffer 0
  {
    unsigned goffA = (unsigned)((((rowBase + aRow) * K) + aCol) * 4);
    async_ld_b128(&As[0][aRow * 20 + aCol], A, goffA);
    async_ld_b128(&As[0][aRow * 20 + aCol + 4], A, goffA + 16);
    unsigned goffB = (unsigned)(((bRow * N) + colBase + bCol) * 4);
    async_ld_b128(&Bs[0][bRow * 80 + bCol], Bm, goffB);
  }

  for (int ck = 0; ck < nck; ++ck) {
    const int cur = ck & 1;
    wait_async0();      // my chunk-ck copies landed
    __syncthreads();    // everyone's landed; prev readers of buf^1 done

    if (ck + 1 < nck) {  // stage chunk ck+1 into the other buffer
      const int k0 = (ck + 1) * 16;
      unsigned goffA = (unsigned)((((rowBase + aRow) * K) + k0 + aCol) * 4);
      async_ld_b128(&As[cur ^ 1][aRow * 20 + aCol], A, goffA);
      async_ld_b128(&As[cur ^ 1][aRow * 20 + aCol + 4], A, goffA + 16);
      unsigned goffB = (unsigned)((((k0 + bRow) * N) + colBase + bCol) * 4);
      async_ld_b128(&Bs[cur ^ 1][bRow * 80 + bCol], Bm, goffB);
    }

    const float* as = As[cur];
    const float* bs = Bs[cur];
#pragma unroll
    for (int kk = 0; kk < 16; kk += 4) {
      v2f a0 = *(const v2f*)(as + (wRow + lh) * 20 + kk + selk);
      v2f a1 = *(const v2f*)(as + (wRow + 16 + lh) * 20 + kk + selk);
      v2f b0, b1;
      b0.x = bs[(kk + selk) * 80 + wCol + lh];
      b0.y = bs[(kk + selk + 1) * 80 + wCol + lh];
      b1.x = bs[(kk + selk) * 80 + wCol + 16 + lh];
      b1.y = bs[(kk + selk + 1) * 80 + wCol + 16 + lh];
      acc[0][0] = wmma4(a0, b0, acc[0][0]);
      acc[0][1] = wmma4(a0, b1, acc[0][1]);
      acc[1][0] = wmma4(a1, b0, acc[1][0]);
      acc[1][1] = wmma4(a1, b1, acc[1][1]);
    }
  }

#pragma unroll
  for (int mi = 0; mi < 2; ++mi)
#pragma unroll
    for (int ni = 0; ni < 2; ++ni)
#pragma unroll
      for (int r = 0; r < 8; ++r) {
        int row = rowBase + wRow + mi * 16 + half * 8 + r;
        int col = colBase + wCol + ni * 16 + lh;
        C[(size_t)row * N + col] = acc[mi][ni][r] + bias[col];
      }
}

// ===========================================================================
// Kernel 2: raw scores S = scale*Q@K^T (causal -inf mask) streamed to attnw,
// with online row max m and sum l (flash-style) saved to workspace.
// grid = (B*H, L/128); 8 waves/block, each wave owns a 16-row query tile.
// K tiles are double-buffered via async LDS copies.
// ===========================================================================
__global__ __launch_bounds__(256) void attn_scores_kernel(
    const float* __restrict__ qkv, float* __restrict__ attnw,
    float* __restrict__ m_ws, float* __restrict__ l_ws) {
  __shared__ float Ks[2][16 * 84];  // K tile [key][hd], stride 84 (16B rows)

  const int tid  = threadIdx.x;
  const int lane = tid & 31;
  const int wid  = tid >> 5;
  const int lh   = lane & 15;
  const int half = lane >> 4;
  const int selk = half * 2;
  const int bh = blockIdx.x;
  const int b  = bh >> 4;
  const int h  = bh & 15;
  const int qt = blockIdx.y * 8 + wid;   // wave's query-tile index (0..63)
  const int qRow0 = qt * 16;
  const float scale = 0.125f;            // 1/sqrt(64)

  // Q fragments resident in registers for the whole key loop (pre-scaled).
  v2f qf[16];
  {
    const float* qbase = qkv + ((size_t)(b * 1024 + qRow0 + lh)) * 3072 + h * 64;
#pragma unroll
    for (int st = 0; st < 16; ++st) {
      v2f v = *(const v2f*)(qbase + st * 4 + selk);
      qf[st] = v * scale;
    }
  }

  float m_run[8], l_run[8];
#pragma unroll
  for (int r = 0; r < 8; ++r) { m_run[r] = NEG_INF; l_run[r] = 0.0f; }

  const int ktMax = blockIdx.y * 8 + 7;  // block-uniform key-tile bound
  const int sRow = tid >> 4;
  const int sCol = (tid & 15) * 4;

  // prologue: stage K tile 0
  {
    unsigned goff =
        (unsigned)((((b * 1024 + sRow) * 3072) + 1024 + h * 64 + sCol) * 4);
    async_ld_b128(&Ks[0][sRow * 84 + sCol], qkv, goff);
  }

  for (int kt = 0; kt <= ktMax; ++kt) {
    const int cur = kt & 1;
    wait_async0();
    __syncthreads();
    if (kt < ktMax) {
      unsigned goff = (unsigned)(
          (((b * 1024 + (kt + 1) * 16 + sRow) * 3072) + 1024 + h * 64 + sCol) *
          4);
      async_ld_b128(&Ks[cur ^ 1][sRow * 84 + sCol], qkv, goff);
    }
    if (kt > qt) continue;  // wave-uniform: EXEC stays all-ones for WMMA

    const float* ks = Ks[cur];
    v8f acc = {};
#pragma unroll
    for (int st = 0; st < 16; ++st) {
      int kk = st * 4 + selk;
      v2f bb = *(const v2f*)(ks + lh * 84 + kk);  // B[k][n] = K[key=n][hd=k]
      acc = wmma4(qf[st], bb, acc);
    }

    const int colg = kt * 16 + lh;
#pragma unroll
    for (int r = 0; r < 8; ++r) {
      int rowg = qRow0 + half * 8 + r;
      float s = acc[r];
      if (colg > rowg) s = NEG_INF;  // causal mask
      attnw[((size_t)bh * 1024 + rowg) * 1024 + colg] = s;  // raw score

      float tmax = s;
      tmax = fmaxf(tmax, __shfl_xor(tmax, 1));
      tmax = fmaxf(tmax, __shfl_xor(tmax, 2));
      tmax = fmaxf(tmax, __shfl_xor(tmax, 4));
      tmax = fmaxf(tmax, __shfl_xor(tmax, 8));
      float newm = fmaxf(m_run[r], tmax);
      float e = __expf(s - newm);
      e += __shfl_xor(e, 1);
      e += __shfl_xor(e, 2);
      e += __shfl_xor(e, 4);
      e += __shfl_xor(e, 8);
      l_run[r] = l_run[r] * __expf(m_run[r] - newm) + e;
      m_run[r] = newm;
    }
  }

  if (lh == 0) {  // lanes 0 and 16 publish row stats
#pragma unroll
    for (int r = 0; r < 8; ++r) {
      int rowg = qRow0 + half * 8 + r;
      m_ws[(size_t)bh * 1024 + rowg] = m_run[r];
      l_ws[(size_t)bh * 1024 + rowg] = l_run[r];
    }
  }
}

// ===========================================================================
// Kernel 3: normalize raw scores -> attn_weights (zeros above diagonal),
// and compute attn_out = P @ V with WMMA via LDS-transposed P tiles.
// V tiles are double-buffered via async LDS copies.
// ===========================================================================
__global__ __launch_bounds__(256) void attn_pv_kernel(
    const float* __restrict__ qkv, float* __restrict__ attnw,
    const float* __restrict__ m_ws, const float* __restrict__ l_ws,
    float* __restrict__ ao) {
  __shared__ float Vs[2][16 * 84];    // V tile [key][hd]
  __shared__ float Ps[8][16 * 18];    // per-wave P tile, A-frag friendly

  const int tid  = threadIdx.x;
  const int lane = tid & 31;
  const int wid  = tid >> 5;
  const int lh   = lane & 15;
  const int half = lane >> 4;
  const int selk = half * 2;
  const int bh = blockIdx.x;
  const int b  = bh >> 4;
  const int h  = bh & 15;
  const int qt = blockIdx.y * 8 + wid;
  const int qRow0 = qt * 16;

  float m_i[8], linv[8];
#pragma unroll
  for (int r = 0; r < 8; ++r) {
    int rowg = qRow0 + half * 8 + r;
    m_i[r] = m_ws[(size_t)bh * 1024 + rowg];
    linv[r] = 1.0f / l_ws[(size_t)bh * 1024 + rowg];
  }

  v8f acc[4] = {};
  const int ktMax = blockIdx.y * 8 + 7;
  const int sRow = tid >> 4;
  const int sCol = (tid & 15) * 4;

  {  // prologue: stage V tile 0
    unsigned goff =
        (unsigned)((((b * 1024 + sRow) * 3072) + 2048 + h * 64 + sCol) * 4);
    async_ld_b128(&Vs[0][sRow * 84 + sCol], qkv, goff);
  }

  for (int kt = 0; kt <= ktMax; ++kt) {
    const int cur = kt & 1;
    wait_async0();
    __syncthreads();
    if (kt < ktMax) {
      unsigned goff = (unsigned)(
          (((b * 1024 + (kt + 1) * 16 + sRow) * 3072) + 2048 + h * 64 + sCol) *
          4);
      async_ld_b128(&Vs[cur ^ 1][sRow * 84 + sCol], qkv, goff);
    }
    if (kt > qt) continue;

    // read raw S, softmax-normalize, write weights + stage P into LDS
    const int colg = kt * 16 + lh;
#pragma unroll
    for (int r = 0; r < 8; ++r) {
      int rowg = qRow0 + half * 8 + r;
      size_t idx = ((size_t)bh * 1024 + rowg) * 1024 + colg;
      float s = attnw[idx];
      float p = __expf(s - m_i[r]) * linv[r];  // masked (-inf) -> exactly 0
      attnw[idx] = p;
      Ps[wid][(half * 8 + r) * 18 + lh] = p;
    }

    // acc(16x64) += P(16x16) @ V(16x64); 4 N-tiles x 4 k-steps
    const float* vs = Vs[cur];
#pragma unroll
    for (int ni = 0; ni < 4; ++ni) {
#pragma unroll
      for (int st = 0; st < 4; ++st) {
        int kk = st * 4 + selk;
        v2f a = *(const v2f*)&Ps[wid][lh * 18 + kk];
        v2f bb;
        bb.x = vs[kk * 84 + ni * 16 + lh];
        bb.y = vs[(kk + 1) * 84 + ni * 16 + lh];
        acc[ni] = wmma4(a, bb, acc[ni]);
      }
    }
  }

  // exact zeros strictly above the diagonal tiles
  for (int kt = qt + 1; kt < 64; ++kt) {
#pragma unroll
    for (int r = 0; r < 8; ++r) {
      int rowg = qRow0 + half * 8 + r;
      attnw[((size_t)bh * 1024 + rowg) * 1024 + kt * 16 + lh] = 0.0f;
    }
  }

  // attn_out in (B, L, H*Hd) layout for the final GEMM
#pragma unroll
  for (int ni = 0; ni < 4; ++ni)
#pragma unroll
    for (int r = 0; r < 8; ++r) {
      int rowg = qRow0 + half * 8 + r;
      ao[((size_t)b * 1024 + rowg) * 1024 + h * 64 + ni * 16 + lh] = acc[ni][r];
    }
}

// ===========================================================================
extern "C" void kernel_launch(void* const* d_in, const int* in_sizes, int n_in,
                              void* d_out, int out_size, void* d_ws, size_t ws_size,
                              hipStream_t stream) {
  (void)in_sizes; (void)n_in; (void)out_size; (void)ws_size;
  const float* x     = (const float*)d_in[0];  // (8,1024,1024)
  const float* W_qkv = (const float*)d_in[1];  // (1024,3072)
  const float* b_qkv = (const float*)d_in[2];  // (3072)
  const float* W_out = (const float*)d_in[3];  // (1024,1024)
  const float* b_out = (const float*)d_in[4];  // (1024)

  float* out   = (float*)d_out;          // 8*1024*1024
  float* attnw = out + 8388608;          // 8*16*1024*1024

  float* qkv_ws = (float*)d_ws;          // 8192 x 3072
  float* ao_ws  = qkv_ws + 25165824;     // 8192 x 1024
  float* m_ws   = ao_ws + 8388608;       // 8*16*1024
  float* l_ws   = m_ws + 131072;         // 8*16*1024

  // 1) qkv = x @ W_qkv + b_qkv          (8192 x 3072 x 1024)
  gemm_bias_wmma<<<dim3(3072 / 64, 8192 / 128), 256, 0, stream>>>(
      x, W_qkv, b_qkv, qkv_ws, 8192, 3072, 1024);

  // 2) raw causal scores + online softmax stats
  attn_scores_kernel<<<dim3(128, 8), 256, 0, stream>>>(qkv_ws, attnw, m_ws, l_ws);

  // 3) normalize weights + P @ V
  attn_pv_kernel<<<dim3(128, 8), 256, 0, stream>>>(qkv_ws, attnw, m_ws, l_ws, ao_ws);

  // 4) output = attn_out @ W_out + b_out (8192 x 1024 x 1024)
  gemm_bias_wmma<<<dim3(1024 / 64, 8192 / 128), 256, 0, stream>>>(
      ao_ws, W_out, b_out, out, 8192, 1024, 1024);
}